// Recommender_27255862461048
// MI455X (gfx1250) — compile-verified
//
#include <hip/hip_runtime.h>
#include <math.h>

#define N_USERS 100000
#define N_ITEMS 50000
#define N_EDGES 2000000
#define DIM     64
#define N_LAYERS 3
#define N_BATCH 4096
#define N_ROWS  (N_USERS + N_ITEMS)      // 150000
#define OUT_DIM ((N_LAYERS + 1) * DIM)   // 256
#define N_TILES (N_ROWS / 16)            // 9375 (exact)
#define WT_STRIDE 66                     // padded (even) row stride for transposed W in LDS

typedef __attribute__((ext_vector_type(2))) float v2f;
typedef __attribute__((ext_vector_type(8))) float v8f;

// ---------------------------------------------------------------- zero agg
__global__ void zero_f4(float4* __restrict__ p, int n4) {
    int i = blockIdx.x * blockDim.x + threadIdx.x;
    if (i < n4) p[i] = make_float4(0.f, 0.f, 0.f, 0.f);
}

// ---------------------------------------------------------------- edge scatter
// one wave32 per edge; lane handles dims [2*lane, 2*lane+1]
__global__ void scatter_edges(const float* __restrict__ user_emb,
                              const float* __restrict__ item_emb,
                              const float* __restrict__ edge_val,
                              const int*   __restrict__ edge_row,
                              const int*   __restrict__ edge_col,
                              float*       __restrict__ agg) {
    int wave = (blockIdx.x * blockDim.x + threadIdx.x) >> 5;
    int lane = threadIdx.x & 31;
    if (wave >= N_EDGES) return;
    int   r = edge_row[wave];
    int   c = edge_col[wave];
    float v = edge_val[wave];
    float2 iv = ((const float2*)(item_emb + (size_t)c * DIM))[lane];
    float2 uv = ((const float2*)(user_emb + (size_t)r * DIM))[lane];
    float* du = agg + (size_t)r * DIM + lane * 2;                 // agg_u
    float* di = agg + (size_t)(N_USERS + c) * DIM + lane * 2;     // agg_i
    atomicAdd(du,     v * iv.x);
    atomicAdd(du + 1, v * iv.y);
    atomicAdd(di,     v * uv.x);
    atomicAdd(di + 1, v * uv.y);
}

// ---------------------------------------------------------------- GEMM + act + l2norm + layer0 copy
// block = 128 threads = 4 waves; wave handles a 16-row tile, all 3 layers.
__global__ __launch_bounds__(128)
void gcn_layers(const float* __restrict__ agg,
                const float* __restrict__ user_emb,
                const float* __restrict__ item_emb,
                const float* __restrict__ W,      // [3][64][64] row-major (k, n)
                const float* __restrict__ bias,   // [3][64]
                float*       __restrict__ out) {  // d_out base (out[0]=loss)
    __shared__ float sWt[N_LAYERS * DIM * WT_STRIDE]; // transposed: sWt[l][n][k], ~50.7 KB
    __shared__ float sB[N_LAYERS * DIM];
    __shared__ float sA[64 * DIM];                    // 16 KB: 4 tiles x 16 rows

    const int tid = threadIdx.x;
    // load W transposed: coalesced global reads; padded stride -> ~2-way LDS write conflicts
    for (int i = tid; i < N_LAYERS * DIM * DIM; i += 128) {
        int l   = i >> 12;
        int rem = i & 4095;
        int k   = rem >> 6;
        int n   = rem & 63;
        sWt[l * DIM * WT_STRIDE + n * WT_STRIDE + k] = W[i];
    }
    for (int i = tid; i < N_LAYERS * DIM; i += 128) sB[i] = bias[i];
    const int rowBlock = blockIdx.x * 64;
    for (int i = tid; i < 64 * DIM; i += 128) {
        int r = rowBlock + (i >> 6);
        sA[i] = (r < N_ROWS) ? agg[(size_t)r * DIM + (i & 63)] : 0.0f;
    }
    __syncthreads();

    const int wi   = tid >> 5;
    const int lane = tid & 31;
    const int h    = lane >> 4;   // half-wave
    const int nl   = lane & 15;
    const int tile = blockIdx.x * 4 + wi;
    if (tile >= N_TILES) return;  // whole wave drops -> EXEC all-1s for WMMA
    const int rbase = tile * 16;

    // ---- layer 0: copy original embeddings into out columns [0,64)
    for (int rr = 0; rr < 16; ++rr) {
        int r = rbase + rr;
        const float* src = (r < N_USERS) ? (user_emb + (size_t)r * DIM)
                                         : (item_emb + (size_t)(r - N_USERS) * DIM);
        size_t ob = (r < N_USERS)
                        ? (1 + (size_t)r * OUT_DIM)
                        : (1 + (size_t)N_USERS * OUT_DIM + (size_t)(r - N_USERS) * OUT_DIM);
        float2 val = ((const float2*)src)[lane];
        out[ob + lane * 2]     = val.x;   // out base is +1 float -> scalar (4B-aligned) stores
        out[ob + lane * 2 + 1] = val.y;
    }

    // ---- A fragments for the whole K=64 strip.
    // f32 16x4 A layout: lanes 0-15 & 16-31 both hold M=0..15;
    // VGPR0 = K = kbase + 2h, VGPR1 = K = kbase + 2h + 1  -> contiguous pair, ds_load_b64.
    v2f a[16];
    const float* aRow = sA + ((wi * 16 + nl) * DIM);
#pragma unroll
    for (int s = 0; s < 16; ++s) {
        int k = s * 4 + h * 2;
        a[s] = *(const v2f*)(aRow + k);
    }

    for (int l = 0; l < N_LAYERS; ++l) {
        const float* Wl = sWt + l * DIM * WT_STRIDE;
        v8f acc[4];
#pragma unroll
        for (int t = 0; t < 4; ++t) {
            const float* bcol = Wl + (t * 16 + nl) * WT_STRIDE;  // column n, contiguous in k
            float bn = sB[l * DIM + t * 16 + nl];
            v8f c;
#pragma unroll
            for (int v = 0; v < 8; ++v) c[v] = bn;
#pragma unroll
            for (int s = 0; s < 16; ++s) {
                int k = s * 4 + h * 2;
                v2f bv = *(const v2f*)(bcol + k);   // single ds_load_b64 into a register pair
                c = __builtin_amdgcn_wmma_f32_16x16x4_f32(
                        /*neg_a*/false, a[s], /*neg_b*/false, bv,
                        /*c_mod*/(short)0, c, /*reuse_a*/false, /*reuse_b*/false);
            }
            acc[t] = c;
        }

        // leaky_relu(0.2) + per-row sum of squares.
        // C/D layout: VGPR v -> row (v + 8h), lane&15 -> column (t*16+nl).
        float x[4][8];
        float rs[8];
#pragma unroll
        for (int v = 0; v < 8; ++v) rs[v] = 0.0f;
#pragma unroll
        for (int t = 0; t < 4; ++t)
#pragma unroll
            for (int v = 0; v < 8; ++v) {
                float y = acc[t][v];
                y = (y >= 0.0f) ? y : 0.2f * y;
                x[t][v] = y;
                rs[v] += y * y;
            }
#pragma unroll
        for (int v = 0; v < 8; ++v) {
            float s4 = rs[v];                  // reduce over the 16 lanes of this half
            s4 += __shfl_xor(s4, 1, 32);
            s4 += __shfl_xor(s4, 2, 32);
            s4 += __shfl_xor(s4, 4, 32);
            s4 += __shfl_xor(s4, 8, 32);
            rs[v] = 1.0f / fmaxf(sqrtf(s4), 1e-12f);
        }
#pragma unroll
        for (int v = 0; v < 8; ++v) {
            int r = rbase + v + h * 8;
            size_t ob = (r < N_USERS)
                            ? (1 + (size_t)r * OUT_DIM)
                            : (1 + (size_t)N_USERS * OUT_DIM + (size_t)(r - N_USERS) * OUT_DIM);
            ob += (size_t)(l + 1) * DIM + nl;
#pragma unroll
            for (int t = 0; t < 4; ++t)
                out[ob + t * 16] = x[t][v] * rs[v];
        }
    }
}

// ---------------------------------------------------------------- BPR loss partials (one wave per sample)
__global__ void bpr_partials(const float* __restrict__ out,
                             const int* __restrict__ bu,
                             const int* __restrict__ bp,
                             const int* __restrict__ bn,
                             float* __restrict__ mf_part,
                             float* __restrict__ reg_part) {
    int wave = (blockIdx.x * blockDim.x + threadIdx.x) >> 5;
    int lane = threadIdx.x & 31;
    if (wave >= N_BATCH) return;
    const float* urow = out + 1 + (size_t)bu[wave] * OUT_DIM;
    const float* prow = out + 1 + (size_t)N_USERS * OUT_DIM + (size_t)bp[wave] * OUT_DIM;
    const float* nrow = out + 1 + (size_t)N_USERS * OUT_DIM + (size_t)bn[wave] * OUT_DIM;
    float pos = 0.f, neg = 0.f, reg = 0.f;
    for (int j = lane; j < OUT_DIM; j += 32) {
        float u = urow[j], p = prow[j], n = nrow[j];
        pos += u * p;
        neg += u * n;
        reg += u * u + p * p + n * n;
    }
#pragma unroll
    for (int m = 16; m >= 1; m >>= 1) {
        pos += __shfl_xor(pos, m, 32);
        neg += __shfl_xor(neg, m, 32);
        reg += __shfl_xor(reg, m, 32);
    }
    if (lane == 0) {
        float d = pos - neg;
        // -log_sigmoid(d) = softplus(-d), numerically stable
        float sp = (d > 0.0f) ? log1pf(expf(-d)) : (-d + log1pf(expf(d)));
        mf_part[wave]  = sp;
        reg_part[wave] = reg;
    }
}

// ---------------------------------------------------------------- deterministic final reduce
__global__ void bpr_finalize(const float* __restrict__ mf_part,
                             const float* __restrict__ reg_part,
                             float* __restrict__ out) {
    __shared__ float smf[256], srg[256];
    int t = threadIdx.x;
    float mf = 0.f, rg = 0.f;
    for (int i = t; i < N_BATCH; i += 256) { mf += mf_part[i]; rg += reg_part[i]; }
    smf[t] = mf; srg[t] = rg;
    __syncthreads();
    for (int s = 128; s > 0; s >>= 1) {
        if (t < s) { smf[t] += smf[t + s]; srg[t] += srg[t + s]; }
        __syncthreads();
    }
    if (t == 0) {
        float mf_loss  = smf[0] / (float)N_BATCH;
        float emb_loss = 1e-4f * (srg[0] * 0.5f) / (float)N_BATCH;
        out[0] = mf_loss + emb_loss;
    }
}

// ---------------------------------------------------------------- launch
extern "C" void kernel_launch(void* const* d_in, const int* in_sizes, int n_in,
                              void* d_out, int out_size, void* d_ws, size_t ws_size,
                              hipStream_t stream) {
    const float* user_emb = (const float*)d_in[0];
    const float* item_emb = (const float*)d_in[1];
    const float* edge_val = (const float*)d_in[2];
    const float* W        = (const float*)d_in[3];
    const float* b        = (const float*)d_in[4];
    const int*   edge_row = (const int*)d_in[5];
    const int*   edge_col = (const int*)d_in[6];
    const int*   bu       = (const int*)d_in[7];
    const int*   bp       = (const int*)d_in[8];
    const int*   bn       = (const int*)d_in[9];
    float* out = (float*)d_out;

    float* agg      = (float*)d_ws;                       // 150000*64 f32 = 38.4 MB
    float* mf_part  = agg + (size_t)N_ROWS * DIM;
    float* reg_part = mf_part + N_BATCH;

    // 1) zero agg
    int n4 = (N_ROWS * DIM) / 4;
    zero_f4<<<(n4 + 255) / 256, 256, 0, stream>>>((float4*)agg, n4);

    // 2) edge scatter (one wave per edge)
    long scatter_threads = (long)N_EDGES * 32;
    scatter_edges<<<(unsigned)((scatter_threads + 255) / 256), 256, 0, stream>>>(
        user_emb, item_emb, edge_val, edge_row, edge_col, agg);

    // 3) WMMA GEMM + activation + l2norm + layer-0 copy
    gcn_layers<<<(N_TILES + 3) / 4, 128, 0, stream>>>(agg, user_emb, item_emb, W, b, out);

    // 4) BPR loss
    bpr_partials<<<(N_BATCH * 32) / 256, 256, 0, stream>>>(out, bu, bp, bn, mf_part, reg_part);
    bpr_finalize<<<1, 256, 0, stream>>>(mf_part, reg_part, out);
}